// GCN_17532056502864
// MI455X (gfx1250) — compile-verified
//
#include <hip/hip_runtime.h>
#include <hip/hip_bf16.h>

typedef float v2f __attribute__((ext_vector_type(2)));
typedef float v8f __attribute__((ext_vector_type(8)));

#define IN_DIM 256
#define H1 512
#define H2 128

// ---------------- degree / normalization ----------------

__global__ void k_init_deg(float* deg, int n) {
  int i = blockIdx.x * blockDim.x + threadIdx.x;
  if (i < n) deg[i] = 1.0f;  // self-loop contributes 1 to every node's degree
}

__global__ void k_deg_accum(const int* __restrict__ dst, float* __restrict__ deg, int E) {
  int e = blockIdx.x * blockDim.x + threadIdx.x;
  if (e < E) unsafeAtomicAdd(&deg[dst[e]], 1.0f);
}

__global__ void k_deg_to_dis(float* deg, int n) {
  int i = blockIdx.x * blockDim.x + threadIdx.x;
  if (i < n) deg[i] = rsqrtf(deg[i]);  // deg >= 1 always (self loops)
}

// ---------------- fp32 WMMA GEMM: C[M,N] = A[M,K] @ B[K,N] ----------------
// Workgroup = 8 waves. Block (bx, by) covers rows [bx*128, +128) and cols
// [by*64, +64). The K x 64 weight slice is staged once in LDS; each wave then
// runs a 16x64 strip of V_WMMA_F32_16X16X4_F32 fed from conflict-free LDS
// reads (lanes hit consecutive DWORDs) with only the streamed A row on the
// global-memory path. KDIM is compile-time so LDS is statically sized.

template <int KDIM>
__global__ void __launch_bounds__(256)
k_gemm_wmma_f32(const float* __restrict__ A, const float* __restrict__ B,
                float* __restrict__ C, int M, int Nn) {
  __shared__ float Bs[KDIM * 64];

  const int tid = threadIdx.x;
  const int col_base = blockIdx.y * 64;

  // cooperative staging of B[K][col_base .. col_base+63] into LDS (float4)
  for (int idx = tid * 4; idx < KDIM * 64; idx += 256 * 4) {
    const int k = idx >> 6;
    const int c = idx & 63;
    *(float4*)(Bs + idx) = *(const float4*)(B + (size_t)k * Nn + col_base + c);
  }
  __syncthreads();

  const int wave = tid >> 5;
  const int lane = tid & 31;
  const int tm = blockIdx.x * 8 + wave;      // wave-uniform row tile
  const bool valid = (tm * 16 < M);
  const int half = lane >> 4;                // 0 = lanes 0-15, 1 = lanes 16-31
  const int l16  = lane & 15;
  int row = tm * 16 + l16;
  if (!valid) row = l16;                     // clamp; results discarded
  const float* Arow = A + (size_t)row * KDIM;

  v8f acc[4];
#pragma unroll
  for (int t = 0; t < 4; ++t) acc[t] = (v8f){0.f,0.f,0.f,0.f,0.f,0.f,0.f,0.f};

  for (int k0 = 0; k0 < KDIM; k0 += 4) {
    // A 16x4 tile: lanes 0-15 carry K=k0,k0+1 ; lanes 16-31 carry K=k0+2,k0+3
    const float* ap = Arow + k0 + half * 2;
    v2f av; av.x = ap[0]; av.y = ap[1];
    const int kb = (k0 + half * 2) * 64 + l16;
#pragma unroll
    for (int t = 0; t < 4; ++t) {
      v2f bv; bv.x = Bs[kb + t * 16]; bv.y = Bs[kb + 64 + t * 16];
      acc[t] = __builtin_amdgcn_wmma_f32_16x16x4_f32(
          false, av, false, bv, (short)0, acc[t], false, false);
    }
  }

  if (valid) {
    // C/D layout: VGPR j -> M = j + 8*half, N = l16 within tile
    const int orow = tm * 16 + half * 8;
#pragma unroll
    for (int t = 0; t < 4; ++t) {
      float* cp = C + (size_t)orow * Nn + col_base + t * 16 + l16;
#pragma unroll
      for (int j = 0; j < 8; ++j) cp[(size_t)j * Nn] = acc[t][j];
    }
  }
}

// ---------------- aggregation ----------------

// agg[i,f] = bias[f] + dis[i]^2 * h[i,f]   (bias + self-loop edge); F = 1<<fshift
__global__ void k_init_agg(const float* __restrict__ h, const float* __restrict__ dis,
                           const float* __restrict__ bias, float* __restrict__ agg,
                           long long total, int fshift) {
  long long idx = (long long)blockIdx.x * blockDim.x + threadIdx.x;
  if (idx >= total) return;
  int i = (int)(idx >> fshift);
  int f = (int)(idx & ((1 << fshift) - 1));
  float w = dis[i];
  agg[idx] = bias[f] + w * w * h[idx];
}

// one wave per edge: agg[dst] += dis[src]*dis[dst] * h[src]  (float4 loads)
__global__ void k_scatter(const int* __restrict__ src, const int* __restrict__ dst,
                          const float* __restrict__ dis, const float* __restrict__ h,
                          float* __restrict__ agg, int E, int F) {
  int wid  = (blockIdx.x * blockDim.x + threadIdx.x) >> 5;
  int lane = threadIdx.x & 31;
  if (wid >= E) return;
  int s = src[wid], d = dst[wid];
  float w = dis[s] * dis[d];
  const float* hs = h + (size_t)s * F;
  float* ad = agg + (size_t)d * F;
  for (int f = lane * 4; f < F; f += 32 * 4) {
    float4 v = *(const float4*)(hs + f);
    unsafeAtomicAdd(&ad[f + 0], v.x * w);
    unsafeAtomicAdd(&ad[f + 1], v.y * w);
    unsafeAtomicAdd(&ad[f + 2], v.z * w);
    unsafeAtomicAdd(&ad[f + 3], v.w * w);
  }
}

// PReLU (shared scalar a) + dropout (p=0.5, scale 2), in place
__global__ void k_prelu_dropout(float* __restrict__ h,
                                const unsigned char* __restrict__ mask,
                                const float* __restrict__ a_ptr, long long total) {
  long long idx = (long long)blockIdx.x * blockDim.x + threadIdx.x;
  if (idx >= total) return;
  float a = a_ptr[0];
  float v = h[idx];
  v = (v >= 0.0f) ? v : a * v;
  h[idx] = mask[idx] ? v * 2.0f : 0.0f;
}

// ---------------- host-side launch ----------------

extern "C" void kernel_launch(void* const* d_in, const int* in_sizes, int n_in,
                              void* d_out, int out_size, void* d_ws, size_t ws_size,
                              hipStream_t stream) {
  const float* x  = (const float*)d_in[0];
  const int*   ei = (const int*)d_in[1];   // [2, E] flattened
  const float* W1 = (const float*)d_in[2];
  const float* b1 = (const float*)d_in[3];
  const float* W2 = (const float*)d_in[4];
  const float* b2 = (const float*)d_in[5];
  const float* a  = (const float*)d_in[6];
  const unsigned char* m1 = (const unsigned char*)d_in[7];
  const unsigned char* m2 = (const unsigned char*)d_in[8];

  const int n = in_sizes[0] / IN_DIM;      // 50000
  const int E = in_sizes[1] / 2;           // 400000
  const int* src = ei;
  const int* dst = ei + E;

  // workspace layout (floats): dis | h1 | agg1 | h2
  float* ws   = (float*)d_ws;
  float* dis  = ws;
  float* h1   = ws + (((size_t)n + 255) & ~(size_t)255);
  float* agg1 = h1 + (size_t)n * H1;
  float* h2   = agg1 + (size_t)n * H1;
  float* out  = (float*)d_out;

  const int TB = 256;
  const int row_blocks = (n / 16 + 7) / 8;   // 8 row-tiles per workgroup

  // --- normalization coefficients ---
  k_init_deg<<<(n + TB - 1) / TB, TB, 0, stream>>>(dis, n);
  k_deg_accum<<<(E + TB - 1) / TB, TB, 0, stream>>>(dst, dis, E);
  k_deg_to_dis<<<(n + TB - 1) / TB, TB, 0, stream>>>(dis, n);

  // --- layer 1: h1 = x @ W1 ; agg1 = self-loop + b1 ; scatter ; act ---
  k_gemm_wmma_f32<IN_DIM><<<dim3(row_blocks, H1 / 64), TB, 0, stream>>>(x, W1, h1, n, H1);
  {
    long long tot = (long long)n * H1;
    k_init_agg<<<(int)((tot + TB - 1) / TB), TB, 0, stream>>>(h1, dis, b1, agg1, tot, 9);
    k_scatter<<<(E + 7) / 8, TB, 0, stream>>>(src, dst, dis, h1, agg1, E, H1);
    k_prelu_dropout<<<(int)((tot + TB - 1) / TB), TB, 0, stream>>>(agg1, m1, a, tot);
  }

  // --- layer 2: h2 = feat1 @ W2 ; out = self-loop + b2 ; scatter ; act ---
  k_gemm_wmma_f32<H1><<<dim3(row_blocks, H2 / 64), TB, 0, stream>>>(agg1, W2, h2, n, H2);
  {
    long long tot = (long long)n * H2;
    k_init_agg<<<(int)((tot + TB - 1) / TB), TB, 0, stream>>>(h2, dis, b2, out, tot, 7);
    k_scatter<<<(E + 7) / 8, TB, 0, stream>>>(src, dst, dis, h2, out, E, H2);
    k_prelu_dropout<<<(int)((tot + TB - 1) / TB), TB, 0, stream>>>(out, m2, a, tot);
  }
}